// SpatialTransformer3D_26792005992775
// MI455X (gfx1250) — compile-verified
//
#include <hip/hip_runtime.h>

// CDNA5 / gfx1250 trilinear 3D spatial transformer.
//
// Memory strategy (MI455X): input volume (67 MB) < L2 (192 MB) => all 8-corner
// gathers are L2 hits after first touch. Output is written with non-temporal
// stores so the streamed 67 MB of results does not evict the L2-resident
// input. Wave32 mapping: lanes 0-15 = 16 consecutive k @ channel 0,
// lanes 16-31 = same 16 k @ channel 1 -> 128 B contiguous store per wave.
//
// Matrix path: the affine stage sg = P(3x4) * [xn,yn,zn,1] is computed with
// one V_WMMA_F32_16X16X4_F32 per wave. A carries P in rows 0-2 AND 8-10 so
// both lane-halves of the 16x16 f32 C/D layout (lanes 0-15 -> rows 0-7,
// lanes 16-31 -> rows 8-15) see the transform; B columns are the 16 grid
// points. Each lane then reads its own sg_x/sg_y/sg_z from d[0],d[1],d[2].
//
// Addressing: base pointers are block-uniform (live in SGPRs); all per-lane
// offsets are 32-bit element indices so gathers/stores lower to GVS-form
// global_load/store (saddr + 32-bit voffset) instead of 64-bit VGPR adds.

typedef __attribute__((ext_vector_type(2))) float v2f;
typedef __attribute__((ext_vector_type(8))) float v8f;

#define GSTEP (2.0f / 127.0f)   // linspace(-1,1,128) step

__global__ __launch_bounds__(256) void st3d_wmma_kernel(
    const float* __restrict__ images,   // (4,128,128,128,2) fp32
    const float* __restrict__ params,   // (4,12) fp32
    float* __restrict__ out)            // (4,128,128,128,2) fp32
{
    // blockIdx.x encodes (b, i, j); each block does one full k-row (128 k x 2 ch).
    const int bij = blockIdx.x;
    const int b = bij >> 14;
    const int i = (bij >> 7) & 127;
    const int j = bij & 127;

    const int tid  = threadIdx.x;
    const int lane = tid & 31;
    const int wave = tid >> 5;
    const int col  = lane & 15;          // point index within the wave's group
    const int k    = wave * 16 + col;    // resample z index
    const int c    = lane >> 4;          // channel 0 / 1

    // ---- scalar-load the 3x4 affine (b is block-uniform -> s_load) ----
    const float* __restrict__ P = params + b * 12;
    float pa[12];
#pragma unroll
    for (int q = 0; q < 12; ++q) pa[q] = P[q];

    // ---- A operand: 16x4 f32. rows 0-2 and 8-10 = P, rest 0.
    // A layout: lanes 0-15 hold M=lane, K=0 (a[0]) / K=1 (a[1]);
    //           lanes 16-31 hold M=lane-16, K=2 (a[0]) / K=3 (a[1]).
    const int  row   = lane & 15;
    const int  rr    = (row >= 8) ? (row - 8) : row;
    const bool valid = (row < 3) || (row >= 8 && row < 11);
    const bool lo    = (lane < 16);
    // branch-free selects (EXEC must stay all-1s for WMMA)
    float cK0 = valid ? ((rr == 0) ? pa[0] : (rr == 1) ? pa[4] : pa[8])  : 0.0f;
    float cK1 = valid ? ((rr == 0) ? pa[1] : (rr == 1) ? pa[5] : pa[9])  : 0.0f;
    float cK2 = valid ? ((rr == 0) ? pa[2] : (rr == 1) ? pa[6] : pa[10]) : 0.0f;
    float cK3 = valid ? ((rr == 0) ? pa[3] : (rr == 1) ? pa[7] : pa[11]) : 0.0f;
    v2f a;
    a[0] = lo ? cK0 : cK2;
    a[1] = lo ? cK1 : cK3;

    // ---- B operand: 4x16, column col = [xn, yn, zn_col, 1].
    // B layout: VGPR0 lanes 0-15 = row K=0, lanes 16-31 = row K=2;
    //           VGPR1 lanes 0-15 = row K=1, lanes 16-31 = row K=3.
    const float xn = fmaf((float)j, GSTEP, -1.0f);
    const float yn = fmaf((float)i, GSTEP, -1.0f);
    const float zn = fmaf((float)k, GSTEP, -1.0f);  // k = wave*16 + (lane&15)
    v2f bv;
    bv[0] = lo ? xn : zn;
    bv[1] = lo ? yn : 1.0f;

    // ---- one WMMA per wave: d rows 0-2 (lanes 0-15) / 8-10 (lanes 16-31)
    // both carry sg for this lane's column.
    v8f acc = {};
    v8f d = __builtin_amdgcn_wmma_f32_16x16x4_f32(
        /*neg_a=*/false, a, /*neg_b=*/false, bv,
        /*c_mod=*/(short)0, acc, /*reuse_a=*/false, /*reuse_b=*/false);

    // normalized [-1,1] -> voxel coords (scaled by full dim, as in reference)
    const float x = (d[0] + 1.0f) * 64.0f;   // 0.5*(sg+1)*128
    const float y = (d[1] + 1.0f) * 64.0f;
    const float z = (d[2] + 1.0f) * 64.0f;

    // truncate-toward-zero like astype(int32), then clamp (matches reference)
    const int ix = (int)x, iy = (int)y, iz = (int)z;
    const int x0 = min(max(ix, 0), 127), x1 = min(max(ix + 1, 0), 127);
    const int y0 = min(max(iy, 0), 127), y1 = min(max(iy + 1, 0), 127);
    const int z0 = min(max(iz, 0), 127), z1 = min(max(iz + 1, 0), 127);

    const float wx0 = (float)x1 - x, wx1 = x - (float)x0;
    const float wy0 = (float)y1 - y, wy1 = y - (float)y0;
    const float wz0 = (float)z1 - z, wz1 = z - (float)z0;

    // gather base: flat index = (z*H*W + y*W + x)*C + c  (reference convention)
    // base pointer is block-uniform (SGPR); per-lane 32-bit element offsets.
    const float* __restrict__ V = images + (unsigned)b * (128u * 128u * 128u * 2u);
    const unsigned ry0 = (unsigned)y0 << 7,  ry1 = (unsigned)y1 << 7;   // y * 128
    const unsigned sz0 = (unsigned)z0 << 14, sz1 = (unsigned)z1 << 14;  // z * 128*128
    const unsigned ux0 = (unsigned)x0, ux1 = (unsigned)x1;
    const unsigned uc  = (unsigned)c;

    const unsigned e000 = ((sz0 + ry0 + ux0) << 1) + uc;
    const unsigned e100 = ((sz1 + ry0 + ux0) << 1) + uc;
    const unsigned e010 = ((sz0 + ry1 + ux0) << 1) + uc;
    const unsigned e110 = ((sz1 + ry1 + ux0) << 1) + uc;
    const unsigned e001 = ((sz0 + ry0 + ux1) << 1) + uc;
    const unsigned e101 = ((sz1 + ry0 + ux1) << 1) + uc;
    const unsigned e011 = ((sz0 + ry1 + ux1) << 1) + uc;
    const unsigned e111 = ((sz1 + ry1 + ux1) << 1) + uc;

    // issue all 8 gathers (L2-resident volume), then accumulate
    const float v000 = V[e000];
    const float v100 = V[e100];
    const float v010 = V[e010];
    const float v110 = V[e110];
    const float v001 = V[e001];
    const float v101 = V[e101];
    const float v011 = V[e011];
    const float v111 = V[e111];

    float r =       wx0 * wy0 * wz0 * v000;
    r = fmaf(wx0 * wy0 * wz1, v100, r);
    r = fmaf(wx0 * wy1 * wz0, v010, r);
    r = fmaf(wx0 * wy1 * wz1, v110, r);
    r = fmaf(wx1 * wy0 * wz0, v001, r);
    r = fmaf(wx1 * wy0 * wz1, v101, r);
    r = fmaf(wx1 * wy1 * wz0, v011, r);
    r = fmaf(wx1 * wy1 * wz1, v111, r);

    // non-temporal store: keep L2 free for the gather-resident input volume.
    // Base is block-uniform; per-lane 32-bit offset -> saddr-form store.
    float* __restrict__ orow = out + (unsigned)bij * 256u;
    __builtin_nontemporal_store(r, orow + ((unsigned)k << 1) + uc);
}

extern "C" void kernel_launch(void* const* d_in, const int* in_sizes, int n_in,
                              void* d_out, int out_size, void* d_ws, size_t ws_size,
                              hipStream_t stream) {
    const float* images = (const float*)d_in[0];   // 4*128*128*128*2 fp32
    const float* params = (const float*)d_in[1];   // 4*12 fp32
    float* out = (float*)d_out;                    // 4*128*128*128*2 fp32

    // one block per (b,i,j) row: 4*128*128 blocks, 256 threads (8 waves) each
    dim3 grid(4 * 128 * 128);
    st3d_wmma_kernel<<<grid, 256, 0, stream>>>(images, params, out);
}